// VectorQuantizer_53154515256217
// MI455X (gfx1250) — compile-verified
//
#include <hip/hip_runtime.h>

typedef __attribute__((ext_vector_type(2))) float v2f;
typedef __attribute__((ext_vector_type(8))) float v8f;

#define E_DIM 64
#define N_E   1024
#define NROWS 32768                       // 32*32*32
#define WAVES_PER_BLOCK 8
#define ROWS_PER_WAVE 16
#define ROWS_PER_BLOCK (WAVES_PER_BLOCK * ROWS_PER_WAVE)   // 128
#define NBLOCKS (NROWS / ROWS_PER_BLOCK)                   // 256
#define LDS_FLOATS (N_E * E_DIM + N_E)                     // 66560 floats = 266240 B (<320KB WGP LDS)

__global__ void vq_zero(float* ws) {
    if (threadIdx.x == 0) ws[0] = 0.0f;
}

extern __shared__ float smem[];

__global__ __launch_bounds__(256)
void vq_main(const float* __restrict__ z, const float* __restrict__ cb,
             float* __restrict__ out_zq, float* __restrict__ out_idx,
             float* __restrict__ loss_acc)
{
    float* s_cb = smem;              // [N_E * E_DIM]
    float* s_en = smem + N_E * E_DIM; // [N_E] squared norms

    const int tid = threadIdx.x;

    // ---- Stage full codebook into LDS (256 KB), float4 coalesced ----
    {
        const float4* g4 = (const float4*)cb;
        float4* s4 = (float4*)s_cb;
        #pragma unroll
        for (int i = 0; i < (N_E * E_DIM / 4) / 256; ++i)
            s4[i * 256 + tid] = g4[i * 256 + tid];
    }
    __syncthreads();

    // ---- Per-entry squared norms ||e_n||^2 from LDS ----
    #pragma unroll
    for (int j = 0; j < N_E / 256; ++j) {
        const int e = j * 256 + tid;
        const float4* r4 = (const float4*)(s_cb + e * E_DIM);
        float s = 0.0f;
        #pragma unroll
        for (int k = 0; k < E_DIM / 4; ++k) {
            float4 v = r4[k];
            s += v.x * v.x + v.y * v.y + v.z * v.z + v.w * v.w;
        }
        s_en[e] = s;
    }
    __syncthreads();

    const int lane = tid & 31;
    const int wave = tid >> 5;
    const int lg   = lane & 15;     // position within 16-lane group
    const int hsel = lane >> 4;     // half-wave select
    const int rowBase = (blockIdx.x * WAVES_PER_BLOCK + wave) * ROWS_PER_WAVE;

    // ---- A fragments: 16x64 tile of z. 32-bit A layout (16x4 per step):
    // lanes 0-15 hold M=0..15, K%4 in {0,1}; lanes 16-31 hold M=0..15, K%4 in {2,3}.
    v2f a[16];
    {
        const float* zr = z + (size_t)(rowBase + lg) * E_DIM + hsel * 2;
        #pragma unroll
        for (int kk = 0; kk < 16; ++kk)
            a[kk] = *(const v2f*)(zr + kk * 4);
    }

    float bestD[8];
    int   bestI[8];
    #pragma unroll
    for (int r = 0; r < 8; ++r) { bestD[r] = 3.4e38f; bestI[r] = 0; }

    // ---- Sweep all 1024 codebook entries in 16-column tiles ----
    for (int t = 0; t < N_E / 16; ++t) {
        const int n = t * 16 + lg;           // this lane's column (codebook index)
        const float* br = s_cb + n * E_DIM + hsel * 2;   // B[k][n] = cb[n][k]
        v2f b[16];
        #pragma unroll
        for (int kk = 0; kk < 16; ++kk)
            b[kk] = *(const v2f*)(br + kk * 4);          // ds_load_b64

        v8f acc = {0.f, 0.f, 0.f, 0.f, 0.f, 0.f, 0.f, 0.f};
        #pragma unroll
        for (int kk = 0; kk < 16; ++kk)
            acc = __builtin_amdgcn_wmma_f32_16x16x4_f32(
                false, a[kk], false, b[kk], (short)0, acc, false, false);

        const float en = s_en[n];
        // D layout: lanes 0-15 slot r -> row r; lanes 16-31 slot r -> row 8+r; col = lg
        #pragma unroll
        for (int r = 0; r < 8; ++r) {
            const float d = en - 2.0f * acc[r];   // ||z||^2 omitted (constant per row)
            if (d < bestD[r]) { bestD[r] = d; bestI[r] = n; }
        }
    }

    // ---- Argmin reduction across the 16-lane group (xor masks stay in-half) ----
    #pragma unroll
    for (int m = 8; m >= 1; m >>= 1) {
        #pragma unroll
        for (int r = 0; r < 8; ++r) {
            const float od = __shfl_xor(bestD[r], m, 32);
            const int   oi = __shfl_xor(bestI[r], m, 32);
            if (od < bestD[r] || (od == bestD[r] && oi < bestI[r])) {
                bestD[r] = od; bestI[r] = oi;
            }
        }
    }

    // ---- Epilogue: gather z_q from LDS codebook, write outputs, loss partial ----
    float lossLocal = 0.0f;
    #pragma unroll
    for (int r = 0; r < 8; ++r) {
        const int row = rowBase + hsel * 8 + r;
        const int bi  = bestI[r];
        const float4 cv = *(const float4*)(s_cb + bi * E_DIM + lg * 4);
        const float4 zv = *(const float4*)(z + (size_t)row * E_DIM + lg * 4);
        *(float4*)(out_zq + (size_t)row * E_DIM + lg * 4) = cv;
        const float dx = cv.x - zv.x, dy = cv.y - zv.y;
        const float dz = cv.z - zv.z, dw = cv.w - zv.w;
        lossLocal += dx * dx + dy * dy + dz * dz + dw * dw;
        if (lg == 0) out_idx[row] = (float)bi;
    }

    #pragma unroll
    for (int m = 16; m >= 1; m >>= 1)
        lossLocal += __shfl_xor(lossLocal, m, 32);
    if (lane == 0) atomicAdd(loss_acc, lossLocal);
}

__global__ void vq_fin(const float* ws, float* out_loss) {
    if (threadIdx.x == 0)
        out_loss[0] = 1.25f * ws[0] / (float)((size_t)NROWS * E_DIM);
}

extern "C" void kernel_launch(void* const* d_in, const int* in_sizes, int n_in,
                              void* d_out, int out_size, void* d_ws, size_t ws_size,
                              hipStream_t stream) {
    const float* z  = (const float*)d_in[0];   // [32,32,32,64] f32
    const float* cb = (const float*)d_in[1];   // [1024,64] f32

    float* out      = (float*)d_out;
    float* out_zq   = out;                          // 2097152 floats
    float* out_idx  = out + (size_t)NROWS * E_DIM;  // 32768 floats
    float* out_loss = out_idx + NROWS;              // 1 float
    float* ws       = (float*)d_ws;

    vq_zero<<<1, 32, 0, stream>>>(ws);
    const size_t ldsBytes = (size_t)LDS_FLOATS * sizeof(float);
    vq_main<<<NBLOCKS, 256, ldsBytes, stream>>>(z, cb, out_zq, out_idx, ws);
    vq_fin<<<1, 32, 0, stream>>>(ws, out_loss);
}